// AttentionAggregator_50852412785041
// MI455X (gfx1250) — compile-verified
//
#include <hip/hip_runtime.h>
#include <hip/hip_bf16.h>

typedef __attribute__((ext_vector_type(2))) float v2f;
typedef __attribute__((ext_vector_type(8))) float v8f;

#define FEAT      128
#define K_NEIGH   10
#define ROWS      16          // output rows per workgroup (one 16xN WMMA tile)
#define CDIM      256         // concat dim = 2*FEAT
#define LDS_PAD   4
#define LDS_STRIDE (CDIM + LDS_PAD)   // 260 floats -> bank-conflict-free A frags

__launch_bounds__(32)
__global__ void gat_attention_agg_wmma(const int*   __restrict__ nodes,
                                       const int*   __restrict__ neigh_idx,
                                       const float* __restrict__ self_emb,
                                       const float* __restrict__ neigh_emb,
                                       const float* __restrict__ weight,   // [128][256]
                                       const float* __restrict__ alpha,    // [256]
                                       float*       __restrict__ out,      // [B][128]
                                       int B)
{
    __shared__ float comb[ROWS * LDS_STRIDE];   // combined = [self_f | agg], 16x256 (+pad)

    const int lane = threadIdx.x;               // 0..31 (wave32)
    const int b0   = blockIdx.x * ROWS;
    const bool full_tile = (b0 + ROWS) <= B;    // uniform: whole 16-row tile in range

    // Per-lane alpha slices: lane covers 4 contiguous channels of the 128-dim dot.
    const float4 a_self  = *(const float4*)(alpha + lane * 4);
    const float4 a_neigh = *(const float4*)(alpha + FEAT + lane * 4);

    // ---------------- Phase 1: gather + attention (whole wave per row) -----------
    for (int r = 0; r < ROWS; ++r) {
        const int b = b0 + r;
        float4 sf = make_float4(0.f, 0.f, 0.f, 0.f);
        float4 ag = make_float4(0.f, 0.f, 0.f, 0.f);

        if (full_tile || b < B) {               // uniform branch
            const int node = nodes[b];
            sf = *(const float4*)(self_emb + (size_t)node * FEAT + lane * 4);

            float p = sf.x * a_self.x + sf.y * a_self.y +
                      sf.z * a_self.z + sf.w * a_self.w;
            #pragma unroll
            for (int off = 16; off >= 1; off >>= 1)
                p += __shfl_xor(p, off, 32);
            const float s_self = p;

            float denom = 0.f;
            for (int k = 0; k < K_NEIGH; ++k) {
                const int ni = neigh_idx[b * K_NEIGH + k];
                const float4 nf =
                    *(const float4*)(neigh_emb + (size_t)ni * FEAT + lane * 4);

                float q = nf.x * a_neigh.x + nf.y * a_neigh.y +
                          nf.z * a_neigh.z + nf.w * a_neigh.w;
                #pragma unroll
                for (int off = 16; off >= 1; off >>= 1)
                    q += __shfl_xor(q, off, 32);

                float s = s_self + q;
                s = s > 0.f ? s : 0.f;          // relu(score)
                const float w = __expf(s);      // unnormalized attention
                denom += w;
                ag.x += w * nf.x; ag.y += w * nf.y;
                ag.z += w * nf.z; ag.w += w * nf.w;
            }
            const float inv = 1.0f / denom;
            ag.x *= inv; ag.y *= inv; ag.z *= inv; ag.w *= inv;
        }

        float* row = comb + r * LDS_STRIDE;
        *(float4*)(row + lane * 4)        = sf;   // combined[:,0:128]   = self_f
        *(float4*)(row + FEAT + lane * 4) = ag;   // combined[:,128:256] = agg
    }
    __syncthreads();   // single-wave WG: lowers to s_nop, keeps DS ordering explicit

    // ---------------- Phase 2: out[16x128] = combined[16x256] @ W^T, fused ReLU --
    // V_WMMA_F32_16X16X4_F32 fragment indexing (ISA 7.12.2, 32-bit A 16x4):
    //   lane L: m = L&15, kb = (L>>4)*2; A vgprs = combined[m][c0+kb], [c0+kb+1]
    //   B (4x16 = W^T chunk): n = L&15;   B vgprs = W[n][c0+kb], W[n][c0+kb+1]
    const int m    = lane & 15;
    const int kb   = (lane >> 4) * 2;
    const int half = (lane >> 4) * 8;           // D rows: vgpr v -> m = v + half
    const float* arow = comb + m * LDS_STRIDE;

    for (int nt = 0; nt < FEAT / 16; ++nt) {    // 8 column tiles of 16
        const int    n    = nt * 16 + m;
        const float* wrow = weight + (size_t)n * CDIM;

        v8f d = {};
        #pragma unroll 8
        for (int c0 = 0; c0 < CDIM; c0 += 4) {
            v2f a, bm;
            a[0]  = arow[c0 + kb];
            a[1]  = arow[c0 + kb + 1];
            bm[0] = wrow[c0 + kb];
            bm[1] = wrow[c0 + kb + 1];
            d = __builtin_amdgcn_wmma_f32_16x16x4_f32(
                    /*neg_a=*/false, a, /*neg_b=*/false, bm,
                    /*c_mod=*/(short)0, d, /*reuse_a=*/false, /*reuse_b=*/false);
        }

        float* outp = out + (size_t)(b0 + half) * FEAT + nt * 16 + m;
        if (full_tile) {
            // Fast path (uniform): unguarded column-strided stores.
            #pragma unroll
            for (int v = 0; v < 8; ++v) {
                float val = d[v];
                val = val > 0.f ? val : 0.f;    // fused final ReLU
                outp[(size_t)v * FEAT] = val;
            }
        } else {
            #pragma unroll
            for (int v = 0; v < 8; ++v) {
                if (b0 + half + v < B) {
                    float val = d[v];
                    val = val > 0.f ? val : 0.f;
                    outp[(size_t)v * FEAT] = val;
                }
            }
        }
    }
}

extern "C" void kernel_launch(void* const* d_in, const int* in_sizes, int n_in,
                              void* d_out, int out_size, void* d_ws, size_t ws_size,
                              hipStream_t stream)
{
    const int*   nodes     = (const int*)  d_in[0];
    const int*   neigh_idx = (const int*)  d_in[1];
    const float* self_emb  = (const float*)d_in[2];
    const float* neigh_emb = (const float*)d_in[3];
    const float* weight    = (const float*)d_in[4];
    const float* alpha     = (const float*)d_in[5];
    float*       out       = (float*)      d_out;

    const int B = in_sizes[0];                  // 65536
    const int grid = (B + ROWS - 1) / ROWS;     // 4096 single-wave workgroups

    gat_attention_agg_wmma<<<grid, 32, 0, stream>>>(
        nodes, neigh_idx, self_emb, neigh_emb, weight, alpha, out, B);
}